// GCNLayer_60421599920302
// MI455X (gfx1250) — compile-verified
//
#include <hip/hip_runtime.h>

// ---------------------------------------------------------------------------
// GCN layer for MI455X (gfx1250, wave32):
//   h = x @ W                           (WMMA f32 16x16x4, exact fp32 math)
//   deg/dinv                            (edge-count atomics + rsqrt)
//   out = h*dinv^2 + b + x              (self-loop + bias + residual init)
//   out[col] += h[row]*dinv[r]*dinv[c]  (1 wave/edge, b128 gather + f32 atomics)
//   BatchNorm (batch stats) + ReLU      (column-parallel stats, in-place)
// Memory-bound: ~2 GB total traffic (edge gather/scatter dominates); h and
// out both fit in the 192 MB L2, so the scatter phase runs at L2 rates.
// GEMM is only 3.28 GFLOP -> keep it exact fp32 on the WMMA path.
// ---------------------------------------------------------------------------

#define N_NODES 100000
#define N_EDGES 1600000
#define D 128
#define WPITCH (D + 4)   // LDS pitch for transposed W: breaks 512B bank stride
#define BN_EPS 1e-5f

typedef float v2f __attribute__((ext_vector_type(2)));
typedef float v8f __attribute__((ext_vector_type(8)));
typedef int   v4i __attribute__((ext_vector_type(4)));

// Async global->LDS staging (CDNA5 ASYNCcnt path), guarded so a missing
// builtin falls back to plain loads instead of breaking the build.
#if defined(__has_builtin)
#  if __has_builtin(__builtin_amdgcn_global_load_async_to_lds_b128) && \
      __has_builtin(__builtin_amdgcn_s_wait_asynccnt)
#    define HAVE_ASYNC_LDS 1
#  endif
#endif
#ifndef HAVE_ASYNC_LDS
#  define HAVE_ASYNC_LDS 0
#endif

// ---------------- degree / dinv --------------------------------------------

__global__ void init_kernel(float* __restrict__ deg, float* __restrict__ sums) {
    int i = blockIdx.x * blockDim.x + threadIdx.x;
    if (i < N_NODES) deg[i] = 1.0f;           // self-loop contributes 1
    if (i < 2 * D)   sums[i] = 0.0f;          // BN sum / sumsq accumulators
}

__global__ void edge_deg_kernel(const long long* __restrict__ col,
                                float* __restrict__ deg) {
    int e = blockIdx.x * blockDim.x + threadIdx.x;
    if (e < N_EDGES) {
        atomicAdd(&deg[(int)col[e]], 1.0f);
    }
}

__global__ void dinv_kernel(float* __restrict__ deg) {
    int i = blockIdx.x * blockDim.x + threadIdx.x;
    if (i < N_NODES) deg[i] = rsqrtf(deg[i]); // deg >= 1, safe
}

// ---------------- GEMM: h = x @ W via V_WMMA_F32_16X16X4_F32 ---------------
// Block: 256 threads = 8 waves; computes a 16x128 strip of h.
// Wave w owns output cols [16w, 16w+16). K-loop: 32 WMMAs of K=4.
// W is staged TRANSPOSED in LDS (sWt[n][k], pitch 132) so both the A and B
// fragments are single aligned ds_load_b64s (two consecutive K values).
// Fragment layouts per CDNA5 ISA 7.12.2 (fp32):
//   A 16x4 : M = lane%16; lane<16 -> {K=0,K=1}, lane>=16 -> {K=2,K=3}
//   B 4x16 : N = lane%16; lane-half selects K pair (v0/v1 = consecutive K)
//   C 16x16: N = lane%16; VGPR j -> M = (lane/16)*8 + j

__global__ __launch_bounds__(256) void gemm_wmma_kernel(
        const float* __restrict__ x, const float* __restrict__ W,
        float* __restrict__ h) {
    __shared__ float sWt[D * WPITCH];  // 66 KB  [n][k] (transposed W)
    __shared__ float sX[16 * D];       //  8 KB  [m][k]

    const int tid  = threadIdx.x;
    const int wave = tid >> 5;
    const int lane = tid & 31;
    const int row0 = blockIdx.x * 16;  // 100000 = 6250 * 16, exact

    // ---- stage x tile (row-major copy): async global->LDS when available
#if HAVE_ASYNC_LDS
    {
        typedef __attribute__((address_space(1))) v4i* g4p;  // global int4*
        typedef __attribute__((address_space(3))) v4i* l4p;  // LDS int4*
        const float* gx = x + (size_t)row0 * D;
        for (int c = tid; c < 16 * D / 4; c += 256) {
            // AS1 keeps the full 64-bit VA; AS3 offset = low 32 bits of the
            // generic LDS address (aperture layout, ISA 10.2).
            g4p gp = (g4p)(unsigned long long)(uintptr_t)(gx + c * 4);
            l4p lp = (l4p)(unsigned int)(uintptr_t)(&sX[c * 4]);
            __builtin_amdgcn_global_load_async_to_lds_b128(gp, lp, 0, 0);
        }
        __builtin_amdgcn_s_wait_asynccnt(0);
    }
#else
    for (int i = tid * 4; i < 16 * D; i += 256 * 4)
        *(float4*)&sX[i] = *(const float4*)&x[(size_t)row0 * D + i];
#endif

    // ---- stage W transposed: coalesced b128 global reads, scatter to LDS
    for (int i = tid * 4; i < D * D; i += 256 * 4) {
        const int k = i / D;          // row of W
        const int n = i % D;          // first of 4 columns
        float4 w4 = *(const float4*)&W[i];
        sWt[(n + 0) * WPITCH + k] = w4.x;
        sWt[(n + 1) * WPITCH + k] = w4.y;
        sWt[(n + 2) * WPITCH + k] = w4.z;
        sWt[(n + 3) * WPITCH + k] = w4.w;
    }
    __syncthreads();

    const int n0    = wave * 16;          // this wave's col tile
    const int mn    = lane & 15;          // M for A-frag, N for B/C frags
    const int khalf = (lane >> 4) * 2;    // 0 or 2: K-pair select

    const float* aRow = &sX[mn * D + khalf];
    const float* bRow = &sWt[(n0 + mn) * WPITCH + khalf];

    v8f acc = {};
    #pragma unroll
    for (int kb = 0; kb < D; kb += 4) {
        v2f a  = *(const v2f*)(aRow + kb);   // one ds_load_b64
        v2f bf = *(const v2f*)(bRow + kb);   // one ds_load_b64
        acc = __builtin_amdgcn_wmma_f32_16x16x4_f32(
                  /*neg_a=*/false, a, /*neg_b=*/false, bf,
                  /*c_mod=*/(short)0, acc, /*reuse_a=*/false, /*reuse_b=*/false);
    }

    const int mbase = (lane >> 4) * 8;
    #pragma unroll
    for (int j = 0; j < 8; ++j)
        h[(size_t)(row0 + mbase + j) * D + n0 + mn] = acc[j];
}

// ---------------- accumulator init: self-loop + bias + residual ------------

__global__ void init_out_kernel(const float* __restrict__ h,
                                const float* __restrict__ x,
                                const float* __restrict__ dinv,
                                const float* __restrict__ b,
                                float* __restrict__ out) {
    int i4 = blockIdx.x * blockDim.x + threadIdx.x;   // float4 index
    if (i4 >= N_NODES * D / 4) return;
    int node = i4 / (D / 4);
    int d4   = (i4 % (D / 4)) * 4;
    float di = dinv[node];
    float sl = di * di;                               // self-loop norm
    float4 hv = *(const float4*)&h[(size_t)i4 * 4];
    float4 xv = *(const float4*)&x[(size_t)i4 * 4];
    float4 bv = *(const float4*)&b[d4];
    float4 o;
    o.x = hv.x * sl + bv.x + xv.x;
    o.y = hv.y * sl + bv.y + xv.y;
    o.z = hv.z * sl + bv.z + xv.z;
    o.w = hv.w * sl + bv.w + xv.w;
    *(float4*)&out[(size_t)i4 * 4] = o;
}

// ---------------- edge aggregation: 1 wave per edge ------------------------
// 32 lanes x float4 = 128 features. Edge index + norms are wave-uniform:
// force e scalar so index/dinv loads go through the scalar path, leaving the
// hot loop as one b128 gather + 4 global_atomic_add_f32 per lane.

__global__ __launch_bounds__(256) void edge_agg_kernel(
        const float* __restrict__ h, const float* __restrict__ dinv,
        const long long* __restrict__ row, const long long* __restrict__ col,
        float* __restrict__ out) {
    int e = __builtin_amdgcn_readfirstlane(
        blockIdx.x * 8 + (int)(threadIdx.x >> 5));    // wave-uniform edge id
    int lane = threadIdx.x & 31;
    if (e >= N_EDGES) return;
    int r = (int)row[e];
    int c = (int)col[e];
    float w = dinv[r] * dinv[c];
    float4 hv = *(const float4*)&h[(size_t)r * D + lane * 4];
    float* dst = &out[(size_t)c * D + lane * 4];
    atomicAdd(dst + 0, hv.x * w);
    atomicAdd(dst + 1, hv.y * w);
    atomicAdd(dst + 2, hv.z * w);
    atomicAdd(dst + 3, hv.w * w);
}

// ---------------- BatchNorm stats: one thread per feature column -----------

__global__ __launch_bounds__(128) void bn_stats_kernel(
        const float* __restrict__ out, float* __restrict__ sums) {
    int d = threadIdx.x;                 // feature index, coalesced reads
    float s = 0.0f, ss = 0.0f;
    for (int r = blockIdx.x; r < N_NODES; r += gridDim.x) {
        float v = out[(size_t)r * D + d];
        s  += v;
        ss += v * v;
    }
    atomicAdd(&sums[d],     s);
    atomicAdd(&sums[D + d], ss);
}

// ---------------- BN apply + ReLU, in place --------------------------------

__global__ void bn_apply_kernel(float* __restrict__ out,
                                const float* __restrict__ sums,
                                const float* __restrict__ gamma,
                                const float* __restrict__ beta) {
    int i = blockIdx.x * blockDim.x + threadIdx.x;
    if (i >= N_NODES * D) return;
    int d = i & (D - 1);
    const float invN = 1.0f / (float)N_NODES;
    float mean = sums[d] * invN;
    float var  = sums[D + d] * invN - mean * mean;   // biased variance
    float v = (out[i] - mean) * rsqrtf(var + BN_EPS) * gamma[d] + beta[d];
    out[i] = fmaxf(v, 0.0f);
}

// ---------------------------------------------------------------------------

extern "C" void kernel_launch(void* const* d_in, const int* in_sizes, int n_in,
                              void* d_out, int out_size, void* d_ws, size_t ws_size,
                              hipStream_t stream) {
    const float*     x     = (const float*)d_in[0];
    const long long* ei    = (const long long*)d_in[1];   // int64 [2, E]
    const float*     W     = (const float*)d_in[2];
    const float*     b     = (const float*)d_in[3];
    const float*     gamma = (const float*)d_in[4];
    const float*     beta  = (const float*)d_in[5];
    float*           out   = (float*)d_out;

    const long long* rowp = ei;             // edge_index[0] = source
    const long long* colp = ei + N_EDGES;   // edge_index[1] = target

    // Workspace layout: h [N*D] | deg/dinv [N] | sums [2*D]
    float* h    = (float*)d_ws;
    float* deg  = h + (size_t)N_NODES * D;
    float* sums = deg + N_NODES;

    init_kernel<<<(N_NODES + 255) / 256, 256, 0, stream>>>(deg, sums);
    edge_deg_kernel<<<(N_EDGES + 255) / 256, 256, 0, stream>>>(colp, deg);
    dinv_kernel<<<(N_NODES + 255) / 256, 256, 0, stream>>>(deg);

    gemm_wmma_kernel<<<N_NODES / 16, 256, 0, stream>>>(x, W, h);

    init_out_kernel<<<(N_NODES * D / 4 + 255) / 256, 256, 0, stream>>>(
        h, x, deg, b, out);

    edge_agg_kernel<<<(N_EDGES * 32) / 256, 256, 0, stream>>>(
        h, deg, rowp, colp, out);

    bn_stats_kernel<<<512, 128, 0, stream>>>(out, sums);

    bn_apply_kernel<<<(N_NODES * D + 255) / 256, 256, 0, stream>>>(
        out, sums, gamma, beta);
}